// StaticChannelWiseNonLin_1159641170454
// MI455X (gfx1250) — compile-verified
//
#include <hip/hip_runtime.h>

// StaticChannelWiseNonLin for MI455X (gfx1250, wave32).
// Memory-bound streaming kernel: async global->LDS double buffering per wave
// (global_load_async_to_lds_b128 + s_wait_asynccnt pipeline), lane == channel
// so all weights live in registers, scalar/dual fp32 body,
// non-temporal output stores. HBM roofline: ~134 MB @ 23.3 TB/s = ~5.75 us.

#define BLOCK 256
#define WAVES_PER_BLOCK 8
#define TILE_FLOATS 512   // per-wave tile: 512 f32 = 2 KB
#define TILE_BYTES 2048
#define NH 10
#define NC 32

typedef float v2f __attribute__((ext_vector_type(2)));

__device__ __forceinline__ void async_load_b128(unsigned lds_off, const void* gaddr) {
  asm volatile("global_load_async_to_lds_b128 %0, %1, off"
               :
               : "v"(lds_off), "v"((unsigned long long)(uintptr_t)gaddr)
               : "memory");
}

__global__ __launch_bounds__(BLOCK) void channelwise_mlp_kernel(
    const float* __restrict__ u, const float* __restrict__ W1,
    const float* __restrict__ B1, const float* __restrict__ W2,
    const float* __restrict__ B2, float* __restrict__ out, int N, int ntiles) {
  // 8 waves * 2 buffers * 512 floats = 32 KB
  __shared__ float smem[WAVES_PER_BLOCK * 2 * TILE_FLOATS];

  const int lane = (int)(threadIdx.x & 31u);
  const int wave_in_blk = (int)(threadIdx.x >> 5);
  const int gwave = (int)blockIdx.x * WAVES_PER_BLOCK + wave_in_blk;
  const int nwaves = (int)gridDim.x * WAVES_PER_BLOCK;

  // Tiles are 512-float aligned and C == 32 == wave width, so the channel of
  // every element this lane touches is exactly `lane`. Weights -> registers,
  // splatted to float2 so the backend can form packed/dual ops where legal.
  v2f w1v[NH], b1v[NH], w2v[NH];
#pragma unroll
  for (int h = 0; h < NH; ++h) {
    const float a = W1[lane * NH + h];
    const float b = B1[lane * NH + h];
    const float c = W2[lane * NH + h];
    w1v[h] = (v2f){a, a};
    b1v[h] = (v2f){b, b};
    w2v[h] = (v2f){c, c};
  }
  const float b2r = B2[lane];

  float* wbuf = &smem[wave_in_blk * (2 * TILE_FLOATS)];
  // Generic pointer to LDS: low 32 bits are the workgroup-relative LDS byte offset.
  const unsigned lds_base = (unsigned)(uintptr_t)wbuf;

  int cnt = 0;
  if (gwave < ntiles) cnt = (ntiles - 1 - gwave) / nwaves + 1;

  // Prologue: put up to 2 tiles in flight (4 async b128 per tile).
#pragma unroll
  for (int j = 0; j < 2; ++j) {
    if (j < cnt) {
      const int t = gwave + j * nwaves;
      const char* g = (const char*)u + (size_t)t * TILE_BYTES + lane * 16;
      const unsigned l = lds_base + (unsigned)(j * TILE_BYTES) + (unsigned)(lane * 16);
#pragma unroll
      for (int k = 0; k < 4; ++k) async_load_b128(l + k * 512u, g + k * 512);
    }
  }

  const v2f zero2 = (v2f){0.0f, 0.0f};

  for (int j = 0; j < cnt; ++j) {
    // Async loads complete in order: <=4 outstanding means tile j landed.
    if (j + 1 < cnt)
      asm volatile("s_wait_asynccnt 4" ::: "memory");
    else
      asm volatile("s_wait_asynccnt 0" ::: "memory");

    const int buf = j & 1;
    const float* lbuf = wbuf + buf * TILE_FLOATS;

    float x[16];
#pragma unroll
    for (int m = 0; m < 16; ++m) x[m] = lbuf[lane + (m << 5)];

    // Ensure the ds reads retired before this buffer is overwritten.
    asm volatile("s_wait_dscnt 0" ::: "memory");

    // Refill this buffer with tile j+2 while we compute tile j.
    if (j + 2 < cnt) {
      const int t2 = gwave + (j + 2) * nwaves;
      const char* g = (const char*)u + (size_t)t2 * TILE_BYTES + lane * 16;
      const unsigned l = lds_base + (unsigned)(buf * TILE_BYTES) + (unsigned)(lane * 16);
#pragma unroll
      for (int k = 0; k < 4; ++k) async_load_b128(l + k * 512u, g + k * 512);
    }

    const int base = (gwave + j * nwaves) * TILE_FLOATS;
#pragma unroll
    for (int p = 0; p < 8; ++p) {
      const v2f xv = (v2f){x[2 * p], x[2 * p + 1]};
      v2f a = (v2f){b2r, b2r};
#pragma unroll
      for (int h = 0; h < NH; ++h) {
        v2f tt = __builtin_elementwise_fma(xv, w1v[h], b1v[h]);
        tt = __builtin_elementwise_max(tt, zero2);
        a = __builtin_elementwise_fma(tt, w2v[h], a);
      }
      // Write-once output stream: non-temporal store policy.
      __builtin_nontemporal_store(a[0], &out[base + lane + ((2 * p) << 5)]);
      __builtin_nontemporal_store(a[1], &out[base + lane + ((2 * p + 1) << 5)]);
    }
  }

  // Tail for N not divisible by TILE_FLOATS (not hit for the reference shape).
  const int done = ntiles * TILE_FLOATS;
  for (int i = done + (int)(blockIdx.x * BLOCK + threadIdx.x); i < N;
       i += (int)(gridDim.x * BLOCK)) {
    const int c = i & (NC - 1);
    const float xv = u[i];
    float a = B2[c];
#pragma unroll
    for (int h = 0; h < NH; ++h) {
      float tt = __builtin_fmaf(xv, W1[c * NH + h], B1[c * NH + h]);
      tt = fmaxf(tt, 0.0f);
      a = __builtin_fmaf(tt, W2[c * NH + h], a);
    }
    __builtin_nontemporal_store(a, &out[i]);
  }
}

extern "C" void kernel_launch(void* const* d_in, const int* in_sizes, int n_in,
                              void* d_out, int out_size, void* d_ws, size_t ws_size,
                              hipStream_t stream) {
  (void)n_in; (void)out_size; (void)d_ws; (void)ws_size;
  const float* u = (const float*)d_in[0];
  const float* W1 = (const float*)d_in[1];
  const float* B1 = (const float*)d_in[2];
  const float* W2 = (const float*)d_in[3];
  const float* B2 = (const float*)d_in[4];
  float* out = (float*)d_out;

  const int N = in_sizes[0];            // 16,777,216 for the reference shape
  const int ntiles = N / TILE_FLOATS;   // 32,768

  // Target ~4 tiles per wave; 1024 blocks -> 8192 waves for the full problem.
  int nblocks = (ntiles + WAVES_PER_BLOCK * 4 - 1) / (WAVES_PER_BLOCK * 4);
  if (nblocks < 1) nblocks = 1;
  if (nblocks > 1024) nblocks = 1024;

  channelwise_mlp_kernel<<<nblocks, BLOCK, 0, stream>>>(u, W1, B1, W2, B2, out,
                                                        N, ntiles);
}